// SimilarityMapModel_16286515987198
// MI455X (gfx1250) — compile-verified
//
#include <hip/hip_runtime.h>
#include <hip/hip_bf16.h>

typedef __attribute__((ext_vector_type(2))) float v2f;
typedef __attribute__((ext_vector_type(8))) float v8f;

#define WDIM 512
#define HDIM 512
#define EDIM 64
#define EPSF 1e-8f

// ---------------------------------------------------------------------------
// Kernel 1: cosine value map via V_WMMA_F32_16X16X4_F32 (exact fp32 path).
// Each wave handles 16 pixels; 16 accumulating WMMAs cover E=64.
// ---------------------------------------------------------------------------
__global__ __launch_bounds__(256)
void value_map_wmma_kernel(const float* __restrict__ mf,
                           const float* __restrict__ qf,
                           float* __restrict__ vm,
                           int whPerBatch) {
    __shared__ float qs[EDIM];

    const int tid  = threadIdx.x;
    const int lane = tid & 31;
    const int wave = tid >> 5;

    const long long pixelBase = ((long long)blockIdx.x * 8 + wave) * 16;
    const int batch = (int)(pixelBase / whPerBatch);

    if (tid < EDIM) qs[tid] = qf[batch * EDIM + tid];
    __syncthreads();

    // Wave-uniform query norm scale (64 LDS broadcasts, negligible).
    float s = 0.f;
#pragma unroll
    for (int j = 0; j < EDIM; ++j) s += qs[j] * qs[j];
    const float qscale = rsqrtf(s + EPSF);

    const float* base = mf + (size_t)pixelBase * EDIM;
    const int p  = lane & 15;   // pixel owned for A-matrix rows
    const int hi = lane >> 4;   // K-half select per ISA A layout

    v8f  c   = {};
    float ssq = 0.f;

#pragma unroll
    for (int kc = 0; kc < EDIM; kc += 4) {
        // A: 16x4 f32 tile. lanes 0-15: K=kc,kc+1 ; lanes 16-31: K=kc+2,kc+3
        v2f a = *(const v2f*)(base + (size_t)p * EDIM + kc + hi * 2);
        ssq += a.x * a.x + a.y * a.y;

        // B: 4x16 f32, normalized query replicated across all 16 columns.
        v2f bq;
        bq.x = qs[kc + hi * 2 + 0] * qscale;
        bq.y = qs[kc + hi * 2 + 1] * qscale;

        c = __builtin_amdgcn_wmma_f32_16x16x4_f32(
                /*neg_a=*/false, a, /*neg_b=*/false, bq,
                /*c_mod=*/(short)0, c, /*reuse_a=*/false, /*reuse_b=*/false);
    }

    // Per-pixel sum of squares: lane p and lane p+16 hold the two halves.
    const float ssqFull = ssq + __shfl_xor(ssq, 16, 32);

    // D layout: VGPR j, lanes 0-15 -> M=j ; lanes 16-31 -> M=8+j (all N equal).
    // Lane L in 0..7 stores pixel L; lane 16+j stores pixel 8+j.
    const int sp = (lane & 7) | ((lane & 16) >> 1);
    const float ssqP = __shfl(ssqFull, sp, 32);

    const int j = lane & 7;
    float v = c[0];
    v = (j == 1) ? c[1] : v;
    v = (j == 2) ? c[2] : v;
    v = (j == 3) ? c[3] : v;
    v = (j == 4) ? c[4] : v;
    v = (j == 5) ? c[5] : v;
    v = (j == 6) ? c[6] : v;
    v = (j == 7) ? c[7] : v;

    if ((lane & 15) < 8) {
        vm[pixelBase + sp] = v * rsqrtf(ssqP + EPSF);
    }
}

// ---------------------------------------------------------------------------
// Kernel 2a: 5-tap sum along contiguous dim (y / h), zero-padded. L2-resident.
// ---------------------------------------------------------------------------
__global__ __launch_bounds__(256)
void row5_kernel(const float* __restrict__ vm, float* __restrict__ rs,
                 long long n) {
    long long i = (long long)blockIdx.x * blockDim.x + threadIdx.x;
    if (i >= n) return;
    const int y = (int)(i & (HDIM - 1));
    float acc = 0.f;
#pragma unroll
    for (int d = -2; d <= 2; ++d) {
        const int yy = y + d;
        if (yy >= 0 && yy < HDIM) acc += vm[i + d];
    }
    rs[i] = acc;
}

// ---------------------------------------------------------------------------
// Kernel 2b: 5-tap sum along x + fused per-batch packed-key argmax.
// key = sortable(score)<<32 | (0xFFFFFFFF - idx)  -> atomicMax == first argmax
// ---------------------------------------------------------------------------
__device__ __forceinline__ unsigned sortable_f32(float f) {
    unsigned u = __float_as_uint(f);
    return (u & 0x80000000u) ? ~u : (u | 0x80000000u);
}

__device__ __forceinline__ unsigned long long shflxor_u64(unsigned long long v,
                                                          int m) {
    unsigned lo = (unsigned)v, hi = (unsigned)(v >> 32);
    lo = __shfl_xor(lo, m, 32);
    hi = __shfl_xor(hi, m, 32);
    return ((unsigned long long)hi << 32) | lo;
}

__global__ __launch_bounds__(256)
void col5_argmax_kernel(const float* __restrict__ rs,
                        unsigned long long* __restrict__ keys) {
    __shared__ unsigned long long red[8];
    const long long i = (long long)blockIdx.x * blockDim.x + threadIdx.x;
    const int idx   = (int)(i & (long long)(WDIM * HDIM - 1));
    const int batch = (int)(i >> 18);            // / (512*512)
    const int x = idx >> 9;

    float acc = 0.f;
#pragma unroll
    for (int d = -2; d <= 2; ++d) {
        const int xx = x + d;
        if (xx >= 0 && xx < WDIM) acc += rs[i + (long long)d * HDIM];
    }

    unsigned long long key =
        ((unsigned long long)sortable_f32(acc) << 32) |
        (unsigned long long)(0xFFFFFFFFu - (unsigned)idx);

#pragma unroll
    for (int m = 1; m < 32; m <<= 1) {
        unsigned long long o = shflxor_u64(key, m);
        if (o > key) key = o;
    }

    const int lane = threadIdx.x & 31;
    const int wave = threadIdx.x >> 5;
    if (lane == 0) red[wave] = key;
    __syncthreads();
    if (threadIdx.x == 0) {
        unsigned long long k = red[0];
#pragma unroll
        for (int w = 1; w < 8; ++w) k = (red[w] > k) ? red[w] : k;
        atomicMax(&keys[batch], k);
    }
}

// ---------------------------------------------------------------------------
// Init + finalize
// ---------------------------------------------------------------------------
__global__ void init_keys_kernel(unsigned long long* keys, int b) {
    int i = threadIdx.x;
    if (i < b) keys[i] = 0ull;
}

__global__ void finalize_kernel(const unsigned long long* __restrict__ keys,
                                float* __restrict__ out, int b) {
    int i = threadIdx.x;
    if (i >= b) return;
    unsigned long long k = keys[i];
    unsigned s = (unsigned)(k >> 32);
    unsigned idx = 0xFFFFFFFFu - (unsigned)k;
    unsigned bits = (s & 0x80000000u) ? (s & 0x7FFFFFFFu) : ~s;
    float score = __uint_as_float(bits);
    out[2 * i + 0] = (float)(idx >> 9);        // xs = idx // h
    out[2 * i + 1] = (float)(idx & (HDIM - 1)); // ys = idx %  h
    out[2 * b + i] = score;
}

// ---------------------------------------------------------------------------
extern "C" void kernel_launch(void* const* d_in, const int* in_sizes, int n_in,
                              void* d_out, int out_size, void* d_ws,
                              size_t ws_size, hipStream_t stream) {
    const float* mf = (const float*)d_in[0];
    const float* qf = (const float*)d_in[1];
    float* out = (float*)d_out;

    const int b = in_sizes[1] / EDIM;                 // 16
    const long long n = (long long)b * WDIM * HDIM;   // 4.19M pixels

    unsigned long long* keys = (unsigned long long*)d_ws;
    float* vm = (float*)((char*)d_ws + 256);
    float* rs = vm + n;

    init_keys_kernel<<<1, 64, 0, stream>>>(keys, b);

    const int waveTiles = (int)(n / 16);   // 16 pixels per wave-tile
    const int blocks1 = waveTiles / 8;     // 8 waves per 256-thread block
    value_map_wmma_kernel<<<blocks1, 256, 0, stream>>>(mf, qf, vm,
                                                       WDIM * HDIM);

    const int blocks2 = (int)((n + 255) / 256);
    row5_kernel<<<blocks2, 256, 0, stream>>>(vm, rs, n);
    col5_argmax_kernel<<<blocks2, 256, 0, stream>>>(rs, keys);

    finalize_kernel<<<1, 64, 0, stream>>>(keys, out, b);
}